// GATBlock_54554674593852
// MI455X (gfx1250) — compile-verified
//
#include <hip/hip_runtime.h>
#include <math.h>

#define NN   50000
#define EE   800000
#define RR   2000
#define DIMM 64
#define HH   4
#define HCC  256
#define NEG_SLOPE 0.2f

typedef float v2f __attribute__((ext_vector_type(2)));
typedef float v8f __attribute__((ext_vector_type(8)));

// ---------------------------------------------------------------------------
// float atomic max via int punning (works for mixed signs, init = -inf)
// ---------------------------------------------------------------------------
__device__ __forceinline__ void atomic_max_f32(float* addr, float val) {
  if (val >= 0.0f) {
    atomicMax((int*)addr, __float_as_int(val));
  } else {
    atomicMin((unsigned int*)addr, (unsigned int)__float_as_int(val));
  }
}

// ---------------------------------------------------------------------------
// Generic fp32 WMMA GEMM:  C[M,NC] = (reluA ? relu(A) : A)[M,K] * B[K,NC] + bias
// Uses V_WMMA_F32_16X16X4_F32.  B staged in LDS once per block (padded rows),
// 16-row A tile in LDS per iteration, 8 waves sweep the column tiles.
// ---------------------------------------------------------------------------
__global__ void gemm_wmma_f32(const float* __restrict__ A,
                              const float* __restrict__ B,
                              const float* __restrict__ bias,
                              float* __restrict__ C,
                              int M, int K, int NC, int reluA) {
  extern __shared__ float smem[];
  const int ldA = K + 2;       // skew to kill DS bank conflicts on A reads
  const int ldB = NC + 16;     // skew so lane groups 0-15 / 16-31 hit disjoint banks
  float* As = smem;            // 16 x ldA
  float* Bs = smem + 16 * ldA; // K x ldB

  const int tid    = threadIdx.x;
  const int lane   = tid & 31;
  const int wave   = tid >> 5;
  const int nwaves = blockDim.x >> 5;

  // stage B (weights) into LDS once
  for (int idx = tid; idx < K * NC; idx += blockDim.x) {
    int k = idx / NC, c = idx - k * NC;
    Bs[k * ldB + c] = B[idx];
  }

  const int rowTiles = (M + 15) >> 4;
  const int colTiles = NC >> 4;
  const int rhalf = lane >> 4;   // 0: K pair {0,1}, 1: K pair {2,3}
  const int rcol  = lane & 15;   // A row (M) / B,C column (N)

  for (int rt = blockIdx.x; rt < rowTiles; rt += gridDim.x) {
    __syncthreads();
    for (int idx = tid; idx < 16 * K; idx += blockDim.x) {
      int r = idx / K, c = idx - r * K;
      int row = (rt << 4) + r;
      float v = (row < M) ? A[(size_t)row * K + c] : 0.0f;
      if (reluA) v = fmaxf(v, 0.0f);
      As[r * ldA + c] = v;
    }
    __syncthreads();

    for (int ct = wave; ct < colTiles; ct += nwaves) {
      const int cb = ct << 4;
      v8f acc;
      const float bv = bias ? bias[cb + rcol] : 0.0f;
#pragma unroll
      for (int v = 0; v < 8; ++v) acc[v] = bv;

      for (int kk = 0; kk < K; kk += 4) {
        const int k0 = kk + (rhalf << 1);
        v2f a, b;
        a.x = As[rcol * ldA + k0];
        a.y = As[rcol * ldA + k0 + 1];
        b.x = Bs[k0 * ldB + cb + rcol];
        b.y = Bs[(k0 + 1) * ldB + cb + rcol];
        acc = __builtin_amdgcn_wmma_f32_16x16x4_f32(
            false, a, false, b, (short)0, acc, false, false);
      }

#pragma unroll
      for (int v = 0; v < 8; ++v) {
        const int row = (rt << 4) + v + (rhalf << 3);
        if (row < M) C[(size_t)row * NC + cb + rcol] = acc[v];
      }
    }
  }
}

// ---------------------------------------------------------------------------
// Pass 1: per-dst sum of edge attrs + degree  (thread per (edge, dim))
// ---------------------------------------------------------------------------
__global__ void edge_attr_sum(const float* __restrict__ relations,
                              const int* __restrict__ dst,
                              const int* __restrict__ rel_idx,
                              float* __restrict__ attr_sum,
                              float* __restrict__ deg) {
  const int gid = blockIdx.x * blockDim.x + threadIdx.x;
  if (gid >= EE * DIMM) return;
  const int e = gid >> 6;
  const int k = gid & 63;
  const int d = dst[e];
  atomicAdd(&attr_sum[(size_t)d * DIMM + k],
            relations[(size_t)rel_idx[e] * DIMM + k]);
  if (k == 0) atomicAdd(&deg[d], 1.0f);
}

__global__ void normalize_loop_attr(float* __restrict__ attr_sum,
                                    const float* __restrict__ deg) {
  const int gid = blockIdx.x * blockDim.x + threadIdx.x;
  if (gid >= NN * DIMM) return;
  attr_sum[gid] = attr_sum[gid] / fmaxf(deg[gid >> 6], 1.0f);
}

__global__ void fill_neg_inf(float* __restrict__ p, int n) {
  const int gid = blockIdx.x * blockDim.x + threadIdx.x;
  if (gid < n) p[gid] = __int_as_float(0xff800000);
}

// ---------------------------------------------------------------------------
// Pass 2: attention logits (wave per edge, 8 dims/lane, head = lane>>3)
// ---------------------------------------------------------------------------
__global__ void alpha_kernel(const float* __restrict__ x_l,
                             const float* __restrict__ x_r,
                             const float* __restrict__ rel_e,
                             const float* __restrict__ e_loop,
                             const float* __restrict__ att,
                             const int* __restrict__ src,
                             const int* __restrict__ dst,
                             const int* __restrict__ rel_idx,
                             float* __restrict__ alpha,
                             float* __restrict__ amax,
                             int nEdgesTotal) {
  const int e = blockIdx.x * (blockDim.x >> 5) + (threadIdx.x >> 5);
  if (e >= nEdgesTotal) return;          // wave-uniform exit
  const int lane = threadIdx.x & 31;
  int s, d;
  const float* evec;
  if (e < EE) {
    s = src[e]; d = dst[e];
    evec = rel_e + (size_t)rel_idx[e] * HCC;
  } else {
    s = e - EE; d = s;
    evec = e_loop + (size_t)s * HCC;
  }
  const int base = lane * 8;
  const float4* xl4 = (const float4*)(x_l + (size_t)s * HCC + base);
  const float4* xr4 = (const float4*)(x_r + (size_t)d * HCC + base);
  const float4* ev4 = (const float4*)(evec + base);
  const float4* at4 = (const float4*)(att + base);
  float partial = 0.0f;
#pragma unroll
  for (int q = 0; q < 2; ++q) {
    const float4 a = xl4[q], b = xr4[q], c = ev4[q], t = at4[q];
    float m;
    m = a.x + b.x + c.x; m = (m >= 0.0f) ? m : NEG_SLOPE * m; partial += m * t.x;
    m = a.y + b.y + c.y; m = (m >= 0.0f) ? m : NEG_SLOPE * m; partial += m * t.y;
    m = a.z + b.z + c.z; m = (m >= 0.0f) ? m : NEG_SLOPE * m; partial += m * t.z;
    m = a.w + b.w + c.w; m = (m >= 0.0f) ? m : NEG_SLOPE * m; partial += m * t.w;
  }
  // reduce within each 8-lane head group (wave32 shuffles)
  partial += __shfl_xor(partial, 1, 32);
  partial += __shfl_xor(partial, 2, 32);
  partial += __shfl_xor(partial, 4, 32);
  if ((lane & 7) == 0) {
    const int h = lane >> 3;
    alpha[(size_t)e * HH + h] = partial;
    atomic_max_f32(&amax[(size_t)d * HH + h], partial);
  }
}

// ---------------------------------------------------------------------------
// Pass 3: stable exp + per-(dst,head) denominator
// ---------------------------------------------------------------------------
__global__ void expdenom_kernel(float* __restrict__ alpha,
                                const float* __restrict__ amax,
                                float* __restrict__ denom,
                                const int* __restrict__ dst,
                                int nEdgesTotal) {
  const int gid = blockIdx.x * blockDim.x + threadIdx.x;
  if (gid >= nEdgesTotal * HH) return;
  const int e = gid >> 2, h = gid & 3;
  const int d = (e < EE) ? dst[e] : (e - EE);
  const float a = expf(alpha[gid] - amax[(size_t)d * HH + h]);
  alpha[gid] = a;
  atomicAdd(&denom[(size_t)d * HH + h], a);
}

// ---------------------------------------------------------------------------
// Pass 4: weighted scatter-add of x_l[src] into out_acc (wave per edge)
// ---------------------------------------------------------------------------
__global__ void aggregate_kernel(const float* __restrict__ x_l,
                                 const float* __restrict__ alpha,
                                 const float* __restrict__ denom,
                                 const int* __restrict__ src,
                                 const int* __restrict__ dst,
                                 float* __restrict__ out_acc,
                                 int nEdgesTotal) {
  const int e = blockIdx.x * (blockDim.x >> 5) + (threadIdx.x >> 5);
  if (e >= nEdgesTotal) return;
  const int lane = threadIdx.x & 31;
  int s, d;
  if (e < EE) { s = src[e]; d = dst[e]; } else { s = e - EE; d = s; }
  const int base = lane * 8;
  const int h = lane >> 3;
  const float w = alpha[(size_t)e * HH + h] / denom[(size_t)d * HH + h];
  const float4* xl4 = (const float4*)(x_l + (size_t)s * HCC + base);
  float* outp = out_acc + (size_t)d * HCC + base;
#pragma unroll
  for (int q = 0; q < 2; ++q) {
    const float4 a = xl4[q];
    atomicAdd(outp + q * 4 + 0, a.x * w);
    atomicAdd(outp + q * 4 + 1, a.y * w);
    atomicAdd(outp + q * 4 + 2, a.z * w);
    atomicAdd(outp + q * 4 + 3, a.w * w);
  }
}

// ---------------------------------------------------------------------------
// Pass 5: mean over heads + bias
// ---------------------------------------------------------------------------
__global__ void finalize_kernel(const float* __restrict__ out_acc,
                                const float* __restrict__ gnn_bias,
                                float* __restrict__ out) {
  const int gid = blockIdx.x * blockDim.x + threadIdx.x;
  if (gid >= NN * DIMM) return;
  const int n = gid >> 6, k = gid & 63;
  const float* p = out_acc + (size_t)n * HCC + k;
  out[gid] = 0.25f * (p[0] + p[64] + p[128] + p[192]) + gnn_bias[k];
}

// ---------------------------------------------------------------------------
extern "C" void kernel_launch(void* const* d_in, const int* in_sizes, int n_in,
                              void* d_out, int out_size, void* d_ws, size_t ws_size,
                              hipStream_t stream) {
  const float* x          = (const float*)d_in[0];
  const float* relations  = (const float*)d_in[1];
  const float* W_l        = (const float*)d_in[2];
  const float* b_l        = (const float*)d_in[3];
  const float* W_r        = (const float*)d_in[4];
  const float* b_r        = (const float*)d_in[5];
  const float* W_e        = (const float*)d_in[6];
  const float* att        = (const float*)d_in[7];
  const float* gnn_bias   = (const float*)d_in[8];
  const float* W_lin_edge = (const float*)d_in[9];
  const float* b_lin_edge = (const float*)d_in[10];
  const int*   edge_index = (const int*)d_in[11];
  const int*   rel_index  = (const int*)d_in[12];
  const int*   srcIdx = edge_index;
  const int*   dstIdx = edge_index + EE;

  // workspace layout (fp32)
  float* ws = (float*)d_ws;
  size_t o = 0;
  float* deg      = ws + o; o += NN;
  float* attr_sum = ws + o; o += (size_t)NN * DIMM;   // becomes loop_attr in place
  float* x_l      = ws + o; o += (size_t)NN * HCC;
  float* x_r      = ws + o; o += (size_t)NN * HCC;
  float* e_loop   = ws + o; o += (size_t)NN * HCC;    // reused as out_acc later
  float* rel_e    = ws + o; o += (size_t)RR * HCC;
  float* alpha    = ws + o; o += (size_t)(EE + NN) * HH;
  float* amax     = ws + o; o += (size_t)NN * HH;
  float* denom    = ws + o; o += (size_t)NN * HH;

  float* x_out   = (float*)d_out;
  float* rel_out = (float*)d_out + (size_t)NN * DIMM;

  hipMemsetAsync(deg, 0, sizeof(float) * NN, stream);
  hipMemsetAsync(attr_sum, 0, sizeof(float) * (size_t)NN * DIMM, stream);
  hipMemsetAsync(denom, 0, sizeof(float) * (size_t)NN * HH, stream);
  fill_neg_inf<<<(NN * HH + 255) / 256, 256, 0, stream>>>(amax, NN * HH);

  // self-loop attr = mean of incoming edge attrs
  edge_attr_sum<<<(EE * DIMM + 255) / 256, 256, 0, stream>>>(
      relations, dstIdx, rel_index, attr_sum, deg);
  normalize_loop_attr<<<(NN * DIMM + 255) / 256, 256, 0, stream>>>(attr_sum, deg);

  // dense GEMMs on the WMMA fp32 pipe
  auto launch_gemm = [&](const float* A, const float* B, const float* bias,
                         float* C, int M, int K, int NC, int reluA) {
    int rowTiles = (M + 15) / 16;
    int grid = rowTiles < 720 ? rowTiles : 720;
    size_t shmem = (size_t)(16 * (K + 2) + K * (NC + 16)) * sizeof(float);
    gemm_wmma_f32<<<grid, 256, shmem, stream>>>(A, B, bias, C, M, K, NC, reluA);
  };
  launch_gemm(x, W_l, b_l, x_l, NN, DIMM, HCC, 0);
  launch_gemm(x, W_r, b_r, x_r, NN, DIMM, HCC, 0);
  launch_gemm(attr_sum, W_e, nullptr, e_loop, NN, DIMM, HCC, 0);
  launch_gemm(relations, W_e, nullptr, rel_e, RR, DIMM, HCC, 0);
  // relations output path: relu(rel_e) @ W_lin_edge + b  -> second output
  launch_gemm(rel_e, W_lin_edge, b_lin_edge, rel_out, RR, HCC, DIMM, 1);

  const int totalE = EE + NN;
  const int wpb = 256 / 32;  // waves per block
  alpha_kernel<<<(totalE + wpb - 1) / wpb, 256, 0, stream>>>(
      x_l, x_r, rel_e, e_loop, att, srcIdx, dstIdx, rel_index,
      alpha, amax, totalE);
  expdenom_kernel<<<(totalE * HH + 255) / 256, 256, 0, stream>>>(
      alpha, amax, denom, dstIdx, totalE);

  // e_loop is dead now -> reuse as the head-resolved output accumulator
  hipMemsetAsync(e_loop, 0, sizeof(float) * (size_t)NN * HCC, stream);
  aggregate_kernel<<<(totalE + wpb - 1) / wpb, 256, 0, stream>>>(
      x_l, alpha, denom, srcIdx, dstIdx, e_loop, totalE);
  finalize_kernel<<<(NN * DIMM + 255) / 256, 256, 0, stream>>>(
      e_loop, gnn_bias, x_out);
}